// ResGCN_15882789060986
// MI455X (gfx1250) — compile-verified
//
#include <hip/hip_runtime.h>
#include <hip/hip_bf16.h>

#define N_NODES 100000
#define N_EDGES 1600000
#define F_IN    48
#define HID     64
#define BN_EPS  1e-5f

typedef __attribute__((ext_vector_type(2))) float v2f;
typedef __attribute__((ext_vector_type(8))) float v8f;

// ---------------------------------------------------------------------------
// Kernel 1: per-feature batch statistics -> folded BN scale/shift
// scale[c] = bn_w[c] * rsqrt(var+eps) ; shift[c] = bn_b[c] - mean*scale[c]
// One block per feature column, 256 threads grid-stride over rows.
// ---------------------------------------------------------------------------
__global__ void bn_stats_kernel(const float* __restrict__ x,
                                const float* __restrict__ bn_w,
                                const float* __restrict__ bn_b,
                                float* __restrict__ scale,
                                float* __restrict__ shift) {
    __shared__ float s_sum[256];
    __shared__ float s_sq[256];
    const int col = blockIdx.x;            // 0..47
    const int tid = threadIdx.x;
    float s = 0.f, q = 0.f;
    for (int r = tid; r < N_NODES; r += 256) {
        float v = x[r * F_IN + col];
        s += v;
        q += v * v;
    }
    s_sum[tid] = s;
    s_sq[tid]  = q;
    __syncthreads();
    for (int off = 128; off > 0; off >>= 1) {
        if (tid < off) {
            s_sum[tid] += s_sum[tid + off];
            s_sq[tid]  += s_sq[tid + off];
        }
        __syncthreads();
    }
    if (tid == 0) {
        float mean = s_sum[0] * (1.0f / N_NODES);
        float var  = s_sq[0] * (1.0f / N_NODES) - mean * mean;
        float sc   = bn_w[col] * __frsqrt_rn(var + BN_EPS);
        scale[col] = sc;
        shift[col] = bn_b[col] - mean * sc;
    }
}

// ---------------------------------------------------------------------------
// Kernel 2a/2b/2c: degree (with self-loop) -> dinv = rsqrt(deg), in place
// ---------------------------------------------------------------------------
__global__ void deg_init_kernel(float* __restrict__ deg) {
    int i = blockIdx.x * blockDim.x + threadIdx.x;
    if (i < N_NODES) deg[i] = 1.0f;       // self-loop contribution
}

__global__ void deg_count_kernel(const long long* __restrict__ dst,
                                 float* __restrict__ deg) {
    int e = blockIdx.x * blockDim.x + threadIdx.x;
    if (e < N_EDGES) {
        int d = (int)dst[e];
        atomicAdd(&deg[d], 1.0f);
    }
}

__global__ void deg_rsqrt_kernel(float* __restrict__ deg) {
    int i = blockIdx.x * blockDim.x + threadIdx.x;
    if (i < N_NODES) deg[i] = __frsqrt_rn(deg[i]);   // deg >= 1 always
}

// ---------------------------------------------------------------------------
// Kernel 3: fused BN-apply + GEMM via V_WMMA_F32_16X16X4_F32 (wave32).
// Block = 128 threads = 4 waves; block owns 16 rows, each wave owns a 16-wide
// column tile of HID=64. K = 48 stepped by 4 -> 12 WMMA ops per wave.
// Stores x_lin to workspace and seeds out with the self-loop term
// x_lin * dinv^2 (so the edge scatter can accumulate on top).
// ---------------------------------------------------------------------------
__global__ void gemm_wmma_kernel(const float* __restrict__ x,
                                 const float* __restrict__ W,
                                 const float* __restrict__ scale,
                                 const float* __restrict__ shift,
                                 const float* __restrict__ dinv,
                                 float* __restrict__ xlin,
                                 float* __restrict__ out) {
    __shared__ float As[16 * F_IN];     // BN-normalized A tile, row-major
    __shared__ float Ws[F_IN * HID];    // full weight matrix, row-major

    const int tid  = threadIdx.x;       // 0..127
    const int m0   = blockIdx.x * 16;   // 100000 = 16 * 6250, exact

    // Stage A tile with BatchNorm folded in (16*48 = 768 floats).
    for (int i = tid; i < 16 * F_IN; i += 128) {
        int r = i / F_IN, c = i % F_IN;
        As[i] = x[(m0 + r) * F_IN + c] * scale[c] + shift[c];
    }
    // Stage W (48*64 = 3072 floats, stays hot in L2 across blocks).
    for (int i = tid; i < F_IN * HID; i += 128) {
        Ws[i] = W[i];
    }
    __syncthreads();

    const int lane = tid & 31;
    const int wave = tid >> 5;          // 0..3 -> column tile
    const int n0   = wave * 16;
    const int half = lane >> 4;         // 0: lanes 0-15, 1: lanes 16-31
    const int lr   = lane & 15;
    const int koff = half * 2;          // A/B K sub-offset per ISA layout

    v8f acc = {0.f, 0.f, 0.f, 0.f, 0.f, 0.f, 0.f, 0.f};

    #pragma unroll
    for (int k = 0; k < F_IN; k += 4) {
        v2f a, b;
        // A 16x4 f32: lane lr = row M, VGPR0/1 = K = koff / koff+1
        a.x = As[lr * F_IN + k + koff];
        a.y = As[lr * F_IN + k + koff + 1];
        // B 4x16 f32: lane lr = col N, VGPR0/1 = K = koff / koff+1
        b.x = Ws[(k + koff) * HID + n0 + lr];
        b.y = Ws[(k + koff + 1) * HID + n0 + lr];
        acc = __builtin_amdgcn_wmma_f32_16x16x4_f32(
            /*neg_a=*/false, a, /*neg_b=*/false, b,
            /*c_mod=*/(short)0, acc, /*reuse_a=*/false, /*reuse_b=*/false);
    }

    // D layout: VGPR r -> row M = r + 8*half, col N = lr.
    #pragma unroll
    for (int r = 0; r < 8; ++r) {
        int row = m0 + r + 8 * half;
        int col = n0 + lr;
        float v = acc[r];
        float di = dinv[row];
        xlin[row * HID + col] = v;
        out[row * HID + col]  = v * di * di;   // self-loop seed
    }
}

// ---------------------------------------------------------------------------
// Kernel 4: edge scatter. 256 threads = 4 edges/block, lane-in-64 = feature.
// out[dst][h] += x_lin[src][h] * dinv[src] * dinv[dst]  (L2 float atomics)
// ---------------------------------------------------------------------------
__global__ void edge_scatter_kernel(const long long* __restrict__ src,
                                    const long long* __restrict__ dst,
                                    const float* __restrict__ xlin,
                                    const float* __restrict__ dinv,
                                    float* __restrict__ out) {
    int e = blockIdx.x * 4 + (threadIdx.x >> 6);
    int h = threadIdx.x & 63;
    if (e < N_EDGES) {
        int s = (int)src[e];
        int d = (int)dst[e];
        float nrm = dinv[s] * dinv[d];
        atomicAdd(&out[d * HID + h], xlin[s * HID + h] * nrm);
    }
}

// ---------------------------------------------------------------------------
// Kernel 5: out = relu(out + b)
// ---------------------------------------------------------------------------
__global__ void finalize_kernel(const float* __restrict__ b,
                                float* __restrict__ out) {
    int i = blockIdx.x * blockDim.x + threadIdx.x;
    if (i < N_NODES * HID) {
        float v = out[i] + b[i & (HID - 1)];
        out[i] = v > 0.f ? v : 0.f;
    }
}

// ---------------------------------------------------------------------------
extern "C" void kernel_launch(void* const* d_in, const int* in_sizes, int n_in,
                              void* d_out, int out_size, void* d_ws, size_t ws_size,
                              hipStream_t stream) {
    const float*     x    = (const float*)d_in[0];
    const long long* ei   = (const long long*)d_in[1];   // [2, E] int64
    const float*     bn_w = (const float*)d_in[2];
    const float*     bn_b = (const float*)d_in[3];
    const float*     W    = (const float*)d_in[4];
    const float*     bias = (const float*)d_in[5];
    float*           out  = (float*)d_out;

    const long long* src = ei;            // edge_index[0]
    const long long* dst = ei + N_EDGES;  // edge_index[1]

    // Workspace layout (floats): [scale 48 | shift 48 | pad to 128 | dinv N | xlin N*HID]
    float* scale = (float*)d_ws;
    float* shift = scale + 48;
    float* dinv  = scale + 128;
    float* xlin  = dinv + N_NODES;

    // 1) BN statistics -> folded scale/shift
    bn_stats_kernel<<<F_IN, 256, 0, stream>>>(x, bn_w, bn_b, scale, shift);

    // 2) degree with self-loops -> dinv
    deg_init_kernel<<<(N_NODES + 255) / 256, 256, 0, stream>>>(dinv);
    deg_count_kernel<<<(N_EDGES + 255) / 256, 256, 0, stream>>>(dst, dinv);
    deg_rsqrt_kernel<<<(N_NODES + 255) / 256, 256, 0, stream>>>(dinv);

    // 3) BN-apply + GEMM (WMMA) + self-loop seed of out
    gemm_wmma_kernel<<<N_NODES / 16, 128, 0, stream>>>(x, W, scale, shift,
                                                       dinv, xlin, out);

    // 4) edge scatter-add
    edge_scatter_kernel<<<(N_EDGES + 3) / 4, 256, 0, stream>>>(src, dst, xlin,
                                                               dinv, out);

    // 5) bias + relu
    finalize_kernel<<<(N_NODES * HID + 255) / 256, 256, 0, stream>>>(bias, out);
}